// DiffGCN_51359218925816
// MI455X (gfx1250) — compile-verified
//
#include <hip/hip_runtime.h>
#include <hip/hip_bf16.h>

// ---------------------------------------------------------------------------
// DiffGCN on MI455X (gfx1250, wave32).
//   N = 100000 nodes (divisible by 16), H = 32, E = 3.2M edges.
//   Dense GEMMs -> V_WMMA_F32_16X16X4_F32 (f32 WMMA, 16x16 tiles).
//   Edge scatter -> wave-per-edge, lane==feature, native f32 global atomics.
// Workspace layout (floats):
//   hw   @ 0        : N*32   (enc @ W_gcn)
//   agg  @ N*32     : N*32   (scatter accumulator, becomes final h in-place)
//   enc  @ 2*N*32   : N*32   (relu(x@W_enc+b))
//   deg  @ 3*N*32   : N
//   dinv @ 3*N*32+N : N
// ---------------------------------------------------------------------------

#define NN 100000
#define HID 32
#define NTILES (NN / 16)   // 6250, exact

typedef __attribute__((ext_vector_type(2))) float v2f;
typedef __attribute__((ext_vector_type(8))) float v8f;

__device__ __forceinline__ void atomic_add_f32(float* p, float v) {
  // gfx1250 has GLOBAL_ATOMIC_ADD_F32; unsafeAtomicAdd lowers to it
  // (plain atomicAdd(float*) can lower to a CAS loop).
  unsafeAtomicAdd(p, v);
}

// ---- init: agg = 0, deg = 1.0 (self-loop term of GCN degree) ---------------
__global__ void k_init(float* __restrict__ agg, float* __restrict__ deg) {
  int idx = blockIdx.x * blockDim.x + threadIdx.x;
  if (idx < NN * HID) agg[idx] = 0.0f;
  if (idx < NN) deg[idx] = 1.0f;
}

// ---- encoder: enc = relu(x @ W_enc + b_enc), x:[N,7] W:[7,32] --------------
// One wave per 16-row tile; V_WMMA_F32_16X16X4_F32, K padded 7 -> 8 (2 steps).
__global__ void k_encoder(const float* __restrict__ X, const float* __restrict__ W,
                          const float* __restrict__ bias, float* __restrict__ out) {
  const int lane = threadIdx.x & 31;
  const int tile = blockIdx.x * (blockDim.x >> 5) + (threadIdx.x >> 5);
  if (tile >= NTILES) return;                 // wave-uniform, EXEC stays full
  const int half = lane >> 4;                 // A/B K-pair selector
  const int l15  = lane & 15;                 // M (for A) / N (for B,C)
  const int row  = tile * 16 + l15;

  v8f c0 = {}; v8f c1 = {};
#pragma unroll
  for (int s = 0; s < 2; ++s) {
    const int ka = s * 4 + 2 * half;          // lane's K base: {0,1} or {2,3}
    v2f a, b0, b1;
    a.x  = (ka     < 7) ? X[row * 7 + ka]     : 0.0f;
    a.y  = (ka + 1 < 7) ? X[row * 7 + ka + 1] : 0.0f;
    b0.x = (ka     < 7) ? W[ka * HID + l15]            : 0.0f;
    b0.y = (ka + 1 < 7) ? W[(ka + 1) * HID + l15]      : 0.0f;
    b1.x = (ka     < 7) ? W[ka * HID + 16 + l15]       : 0.0f;
    b1.y = (ka + 1 < 7) ? W[(ka + 1) * HID + 16 + l15] : 0.0f;
    c0 = __builtin_amdgcn_wmma_f32_16x16x4_f32(false, a, false, b0, (short)0, c0, false, false);
    c1 = __builtin_amdgcn_wmma_f32_16x16x4_f32(false, a, false, b1, (short)0, c1, false, false);
  }
  const float bc0 = bias[l15], bc1 = bias[16 + l15];
  const int mbase = tile * 16 + 8 * half;     // C layout: M = r + 8*half
#pragma unroll
  for (int r = 0; r < 8; ++r) {
    const int m = mbase + r;
    out[m * HID + l15]      = fmaxf(c0[r] + bc0, 0.0f);
    out[m * HID + 16 + l15] = fmaxf(c1[r] + bc1, 0.0f);
  }
}

// ---- gcn mm: hw = enc @ W_gcn, [N,32]x[32,32], 8 k-steps -------------------
__global__ void k_gcnmm(const float* __restrict__ ENC, const float* __restrict__ W,
                        float* __restrict__ out) {
  const int lane = threadIdx.x & 31;
  const int tile = blockIdx.x * (blockDim.x >> 5) + (threadIdx.x >> 5);
  if (tile >= NTILES) return;
  const int half = lane >> 4;
  const int l15  = lane & 15;
  const int row  = tile * 16 + l15;

  v8f c0 = {}; v8f c1 = {};
#pragma unroll
  for (int s = 0; s < 8; ++s) {
    const int ka = s * 4 + 2 * half;
    v2f a, b0, b1;
    a.x  = ENC[row * HID + ka];
    a.y  = ENC[row * HID + ka + 1];
    b0.x = W[ka * HID + l15];
    b0.y = W[(ka + 1) * HID + l15];
    b1.x = W[ka * HID + 16 + l15];
    b1.y = W[(ka + 1) * HID + 16 + l15];
    c0 = __builtin_amdgcn_wmma_f32_16x16x4_f32(false, a, false, b0, (short)0, c0, false, false);
    c1 = __builtin_amdgcn_wmma_f32_16x16x4_f32(false, a, false, b1, (short)0, c1, false, false);
  }
  const int mbase = tile * 16 + 8 * half;
#pragma unroll
  for (int r = 0; r < 8; ++r) {
    const int m = mbase + r;
    out[m * HID + l15]      = c0[r];
    out[m * HID + 16 + l15] = c1[r];
  }
}

// ---- degree: deg[dst] += 1 per edge ----------------------------------------
__global__ void k_degree(const long long* __restrict__ dst, float* __restrict__ deg, int E) {
  int e = blockIdx.x * blockDim.x + threadIdx.x;
  if (e < E) atomic_add_f32(&deg[(int)dst[e]], 1.0f);
}

// ---- dinv = rsqrt(deg) -----------------------------------------------------
__global__ void k_dinv(const float* __restrict__ deg, float* __restrict__ dinv) {
  int i = blockIdx.x * blockDim.x + threadIdx.x;
  if (i < NN) dinv[i] = rsqrtf(deg[i]);
}

// ---- scatter: agg[dst] += hw[src] * dinv[src]*dinv[dst] --------------------
// One wave32 per edge, lane == feature. agg (12.8 MB) lives in L2.
__global__ void k_aggregate(const float* __restrict__ hw, const float* __restrict__ dinv,
                            const long long* __restrict__ src, const long long* __restrict__ dst,
                            float* __restrict__ agg, int E) {
  const int lane = threadIdx.x & 31;
  const int e = blockIdx.x * (blockDim.x >> 5) + (threadIdx.x >> 5);
  if (e >= E) return;
  const int s = (int)src[e];
  const int d = (int)dst[e];
  const float norm = dinv[s] * dinv[d];
  atomic_add_f32(&agg[d * HID + lane], hw[s * HID + lane] * norm);
}

// ---- finalize (in place on agg): h = relu(agg + hw*dinv^2 + b_gcn) ---------
__global__ void k_finalize(float* __restrict__ agg, const float* __restrict__ hw,
                           const float* __restrict__ dinv, const float* __restrict__ bias) {
  int idx = blockIdx.x * blockDim.x + threadIdx.x;
  if (idx >= NN * HID) return;
  const int i = idx >> 5;
  const int f = idx & 31;
  const float di = dinv[i];
  agg[idx] = fmaxf(agg[idx] + hw[idx] * di * di + bias[f], 0.0f);
}

// ---- edge scorer: sigmoid(<[h[src],h[dst]], W_edge> + b) -------------------
// Wave per edge; lane f handles both halves of the 64-wide dot; xor-shuffle tree.
__global__ void k_edgescore(const float* __restrict__ h, const long long* __restrict__ src,
                            const long long* __restrict__ dst, const float* __restrict__ We,
                            const float* __restrict__ be, float* __restrict__ out, int E) {
  const int lane = threadIdx.x & 31;
  const int e = blockIdx.x * (blockDim.x >> 5) + (threadIdx.x >> 5);
  if (e >= E) return;
  const int s = (int)src[e];
  const int d = (int)dst[e];
  float v = h[s * HID + lane] * We[lane] + h[d * HID + lane] * We[HID + lane];
#pragma unroll
  for (int off = 16; off > 0; off >>= 1) v += __shfl_xor(v, off, 32);
  if (lane == 0) out[e] = 1.0f / (1.0f + __expf(-(v + be[0])));
}

// ---------------------------------------------------------------------------
extern "C" void kernel_launch(void* const* d_in, const int* in_sizes, int n_in,
                              void* d_out, int out_size, void* d_ws, size_t ws_size,
                              hipStream_t stream) {
  const float*     x_t   = (const float*)d_in[0];
  // d_in[1] = x_t_dt: unused by the reference computation
  const long long* eidx  = (const long long*)d_in[2];   // int64 [2, E]
  const float*     W_enc = (const float*)d_in[3];
  const float*     b_enc = (const float*)d_in[4];
  const float*     W_gcn = (const float*)d_in[5];
  const float*     b_gcn = (const float*)d_in[6];
  const float*     W_edge= (const float*)d_in[7];
  const float*     b_edge= (const float*)d_in[8];
  float*           out   = (float*)d_out;

  const int E = in_sizes[2] / 2;                        // 3,200,000
  const long long* src = eidx;
  const long long* dst = eidx + E;

  float* ws   = (float*)d_ws;
  float* hw   = ws;                     // N*32
  float* agg  = ws + (size_t)NN * HID;  // N*32 (becomes h)
  float* enc  = agg + (size_t)NN * HID; // N*32
  float* deg  = enc + (size_t)NN * HID; // N
  float* dinv = deg + NN;               // N

  const int nelem   = NN * HID;                         // 3.2M
  const int gElem   = (nelem + 255) / 256;              // 12500
  const int gTiles  = (NTILES + 7) / 8;                 // 8 waves/block
  const int gEdgeT  = (E + 255) / 256;                  // thread per edge
  const int gEdgeW  = (E + 7) / 8;                      // wave per edge

  k_init     <<<gElem,  256, 0, stream>>>(agg, deg);
  k_encoder  <<<gTiles, 256, 0, stream>>>(x_t, W_enc, b_enc, enc);
  k_gcnmm    <<<gTiles, 256, 0, stream>>>(enc, W_gcn, hw);
  k_degree   <<<gEdgeT, 256, 0, stream>>>(dst, deg, E);
  k_dinv     <<<(NN + 255) / 256, 256, 0, stream>>>(deg, dinv);
  k_aggregate<<<gEdgeW, 256, 0, stream>>>(hw, dinv, src, dst, agg, E);
  k_finalize <<<gElem,  256, 0, stream>>>(agg, hw, dinv, b_gcn);
  k_edgescore<<<gEdgeW, 256, 0, stream>>>(agg, src, dst, W_edge, b_edge, out, E);
}